// GCN_23742579212600
// MI455X (gfx1250) — compile-verified
//
#include <hip/hip_runtime.h>
#include <math.h>

// GCN (3-layer DGL GraphConv, norm='both') for MI455X / gfx1250.
// GEMMs via V_WMMA_F32_16X16X4_F32 (full f32 accuracy; GEMM is not the
// bottleneck -- edge aggregation over L2-resident feature matrices is).

typedef __attribute__((ext_vector_type(2))) float v2f;
typedef __attribute__((ext_vector_type(8))) float v8f;

// ---------------- utility kernels ----------------

__global__ void zero_kernel(float* __restrict__ p, size_t n) {
  size_t i = (size_t)blockIdx.x * blockDim.x + threadIdx.x;
  size_t stride = (size_t)gridDim.x * blockDim.x;
  for (; i < n; i += stride) p[i] = 0.0f;
}

__global__ void degree_kernel(const int* __restrict__ src, const int* __restrict__ dst,
                              int E, float* __restrict__ outdeg, float* __restrict__ indeg) {
  int i = blockIdx.x * blockDim.x + threadIdx.x;
  if (i < E) {
    atomicAdd(&outdeg[src[i]], 1.0f);
    atomicAdd(&indeg[dst[i]], 1.0f);
  }
}

__global__ void norm_kernel(const float* __restrict__ outdeg, const float* __restrict__ indeg,
                            float* __restrict__ outnorm, float* __restrict__ innorm, int N) {
  int i = blockIdx.x * blockDim.x + threadIdx.x;
  if (i < N) {
    outnorm[i] = rsqrtf(fmaxf(outdeg[i], 1.0f));  // clip(deg,1)^-0.5
    innorm[i]  = rsqrtf(fmaxf(indeg[i],  1.0f));
  }
}

// ---------------- WMMA f32 GEMM ----------------
// C[N x M] = (rowscale ? diag(rowscale) : I) * A[N x K] @ B[K x M]
// optional epilogue: C = relu?(C * innorm[row] + bias[col])
// One wave computes one 16x16 tile; wave w of the block owns cols [16w,16w+16).
// Requires N % 16 == 0, K % 4 == 0, M % 16 == 0 (holds: 100000/256/128/64).
template<int K, int M>
__global__ __launch_bounds__(32 * (M / 16))
void gemm_wmma(const float* __restrict__ A, const float* __restrict__ B,
               float* __restrict__ C,
               const float* __restrict__ rowscale,
               const float* __restrict__ innorm, const float* __restrict__ bias,
               int relu) {
  const int wave = threadIdx.x >> 5;
  const int lane = threadIdx.x & 31;
  const int half = lane >> 4;   // 0: lanes 0-15, 1: lanes 16-31
  const int m    = lane & 15;
  const int row0 = blockIdx.x * 16;
  const int col0 = wave * 16;

  const float* __restrict__ Arow = A + (size_t)(row0 + m) * K;
  const float sc = rowscale ? rowscale[row0 + m] : 1.0f;

  v8f acc = {};
#pragma unroll 8
  for (int k = 0; k < K; k += 4) {
    const int ka = k + 2 * half;           // A lane K-pair: {0,1} or {2,3}
    v2f a, b;
    a.x = Arow[ka]     * sc;               // contiguous -> b64 load
    a.y = Arow[ka + 1] * sc;
    b.x = B[(size_t)ka       * M + col0 + m];  // coalesced across lanes
    b.y = B[(size_t)(ka + 1) * M + col0 + m];
    acc = __builtin_amdgcn_wmma_f32_16x16x4_f32(
        /*neg_a=*/false, a, /*neg_b=*/false, b,
        /*c_mod=*/(short)0, acc, /*reuse_a=*/false, /*reuse_b=*/false);
  }

  // C/D layout: vgpr i -> row i + 8*half, col = lane%16
#pragma unroll
  for (int i = 0; i < 8; ++i) {
    const int row = row0 + i + 8 * half;
    const int col = col0 + m;
    float v = acc[i];
    if (innorm) v = v * innorm[row] + bias[col];
    if (relu)   v = fmaxf(v, 0.0f);
    C[(size_t)row * M + col] = v;
  }
}

// ---------------- edge aggregation (segment_sum) ----------------
// agg[dst[e], :] += h[src[e], :] * (srcscale ? srcscale[src[e]] : 1)
// M/4 threads per edge, float4 gathers, f32 atomics into L2-resident agg.
__global__ void aggregate_kernel(const float* __restrict__ h,
                                 const int* __restrict__ src, const int* __restrict__ dst,
                                 const float* __restrict__ srcscale,
                                 float* __restrict__ agg, int E, int M) {
  const int tpe  = M >> 2;                       // threads per edge (32 for M=128)
  const int epb  = blockDim.x / tpe;             // edges per block
  const int e    = blockIdx.x * epb + (int)(threadIdx.x / tpe);
  const int lane = threadIdx.x % tpe;
  if (e >= E) return;
  const int s = src[e], d = dst[e];
  const float sc = srcscale ? srcscale[s] : 1.0f;
  const float4 v = *(const float4*)(h + (size_t)s * M + lane * 4);
  float* out = agg + (size_t)d * M + lane * 4;
  atomicAdd(out + 0, v.x * sc);
  atomicAdd(out + 1, v.y * sc);
  atomicAdd(out + 2, v.z * sc);
  atomicAdd(out + 3, v.w * sc);
}

// out[i,j] = relu?(io[i,j] * innorm[i] + bias[j])   (in place)
__global__ void finish_kernel(float* __restrict__ io, const float* __restrict__ innorm,
                              const float* __restrict__ bias, int N, int M, int relu) {
  const int idx = blockIdx.x * blockDim.x + threadIdx.x;
  if (idx >= N * M) return;
  const int row = idx / M, col = idx % M;
  float v = io[idx] * innorm[row] + bias[col];
  if (relu) v = fmaxf(v, 0.0f);
  io[idx] = v;
}

// ---------------- driver ----------------

extern "C" void kernel_launch(void* const* d_in, const int* in_sizes, int n_in,
                              void* d_out, int out_size, void* d_ws, size_t ws_size,
                              hipStream_t stream) {
  const float* x   = (const float*)d_in[0];
  const int*   src = (const int*)d_in[1];
  const int*   dst = (const int*)d_in[2];
  const float* W1  = (const float*)d_in[3];
  const float* b1  = (const float*)d_in[4];
  const float* W2  = (const float*)d_in[5];
  const float* b2  = (const float*)d_in[6];
  const float* W3  = (const float*)d_in[7];
  const float* b3  = (const float*)d_in[8];
  float* out = (float*)d_out;

  const int IN = 256, HID = 128, CLS = 64;
  const int E = in_sizes[1];
  const int N = in_sizes[0] / IN;          // 100000, divisible by 16

  float* ws      = (float*)d_ws;
  float* outdeg  = ws;
  float* indeg   = ws + (size_t)N;
  float* outnorm = ws + 2 * (size_t)N;
  float* innorm  = ws + 3 * (size_t)N;
  float* bufA    = ws + 4 * (size_t)N;         // N x 128
  float* bufB    = bufA + (size_t)N * HID;     // N x 128

  const int rowtiles = N / 16;

  // ---- degree norms ----
  zero_kernel<<<1024, 256, 0, stream>>>(outdeg, 2 * (size_t)N);
  degree_kernel<<<(E + 255) / 256, 256, 0, stream>>>(src, dst, E, outdeg, indeg);
  norm_kernel<<<(N + 255) / 256, 256, 0, stream>>>(outdeg, indeg, outnorm, innorm, N);

  // ---- layer 1: (x * outnorm) @ W1 -> aggregate -> *innorm + b1 -> relu ----
  gemm_wmma<256, 128><<<rowtiles, 256, 0, stream>>>(x, W1, bufA, outnorm,
                                                    nullptr, nullptr, 0);
  zero_kernel<<<2048, 256, 0, stream>>>(bufB, (size_t)N * HID);
  aggregate_kernel<<<(E + 7) / 8, 256, 0, stream>>>(bufA, src, dst, nullptr, bufB, E, HID);
  finish_kernel<<<(N * HID + 255) / 256, 256, 0, stream>>>(bufB, innorm, b1, N, HID, 1);

  // ---- layer 2: aggregate(h * outnorm) -> @ W2 -> *innorm + b2 -> relu (fused) ----
  zero_kernel<<<2048, 256, 0, stream>>>(bufA, (size_t)N * HID);
  aggregate_kernel<<<(E + 7) / 8, 256, 0, stream>>>(bufB, src, dst, outnorm, bufA, E, HID);
  gemm_wmma<128, 128><<<rowtiles, 256, 0, stream>>>(bufA, W2, bufB, nullptr,
                                                    innorm, b2, 1);

  // ---- layer 3: (h * outnorm) @ W3 -> aggregate into d_out -> *innorm + b3 ----
  gemm_wmma<128, 64><<<rowtiles, 128, 0, stream>>>(bufB, W3, bufA, outnorm,
                                                   nullptr, nullptr, 0);
  zero_kernel<<<1024, 256, 0, stream>>>(out, (size_t)N * CLS);
  aggregate_kernel<<<(E + 15) / 16, 256, 0, stream>>>(bufA, src, dst, nullptr, out, E, CLS);
  finish_kernel<<<(N * CLS + 255) / 256, 256, 0, stream>>>(out, innorm, b3, N, CLS, 0);
}